// GCN_23828478558291
// MI455X (gfx1250) — compile-verified
//
#include <hip/hip_runtime.h>

typedef __attribute__((ext_vector_type(2))) float v2f;
typedef __attribute__((ext_vector_type(8))) float v8f;

#define HDIM 32

// ---------- utility fills ----------
__global__ void gcn_fill_kernel(float* __restrict__ p, float val, int n) {
  int i = blockIdx.x * blockDim.x + threadIdx.x;
  if (i < n) p[i] = val;
}

// ---------- degree: deg[dst] += 1 per edge (deg pre-initialized to 1.0 for self loop) ----
__global__ void gcn_degree_kernel(const int* __restrict__ dst, float* __restrict__ deg, int E) {
  int e = blockIdx.x * blockDim.x + threadIdx.x;
  if (e < E) atomicAdd(&deg[dst[e]], 1.0f);
}

// ---------- dis = rsqrt(deg) in place (deg >= 1 always: self loops) ----------
__global__ void gcn_rsqrt_kernel(float* __restrict__ d, int n) {
  int i = blockIdx.x * blockDim.x + threadIdx.x;
  if (i < n) d[i] = rsqrtf(d[i]);
}

// ---------- h = X @ W   (N x 32) = (N x 32)(32 x 32), f32 WMMA 16x16x4 ----------
// One wave computes a 16-row x 32-col tile: two v8f accumulators, K consumed 4 at a time.
__global__ void gcn_gemm_wmma_kernel(const float* __restrict__ X,
                                     const float* __restrict__ W,
                                     float* __restrict__ H, int N) {
  const int lane = threadIdx.x & 31;
  const int wid  = threadIdx.x >> 5;
  const int tile = blockIdx.x * (blockDim.x >> 5) + wid;
  const int nTiles = (N + 15) >> 4;
  if (tile >= nTiles) return;                 // whole wave exits together (EXEC stays uniform)

  const int half = lane >> 4;                 // 0: lanes 0-15, 1: lanes 16-31
  const int l16  = lane & 15;
  int row = tile * 16 + l16;
  if (row >= N) row = N - 1;                  // clamp loads; WMMA needs EXEC all-ones

  v8f acc0 = {};                              // output cols 0..15
  v8f acc1 = {};                              // output cols 16..31

#pragma unroll
  for (int k = 0; k < HDIM; k += 4) {
    // A fragment (16x4 f32): v0 holds K = k+2*half, v1 holds K = k+2*half+1
    v2f a;
    const int ka = k + 2 * half;
    a.x = X[(size_t)row * HDIM + ka];
    a.y = X[(size_t)row * HDIM + ka + 1];

    // B fragments (4x16 f32), mirrored layout: lane half selects K pair, l16 = column
    v2f b0, b1;
    b0.x = W[ka * HDIM + l16];
    b0.y = W[(ka + 1) * HDIM + l16];
    b1.x = W[ka * HDIM + 16 + l16];
    b1.y = W[(ka + 1) * HDIM + 16 + l16];

    acc0 = __builtin_amdgcn_wmma_f32_16x16x4_f32(false, a, false, b0, (short)0, acc0, false, false);
    acc1 = __builtin_amdgcn_wmma_f32_16x16x4_f32(false, a, false, b1, (short)0, acc1, false, false);
  }

  // C/D layout: VGPR r, lanes 0-15 -> M=r, N=l16 ; lanes 16-31 -> M=8+r
#pragma unroll
  for (int r = 0; r < 8; ++r) {
    const int orow = tile * 16 + r + 8 * half;
    if (orow < N) {
      H[(size_t)orow * HDIM + l16]      = acc0[r];
      H[(size_t)orow * HDIM + 16 + l16] = acc1[r];
    }
  }
}

// ---------- per-edge scatter: agg[dst] += h[src] * dis[src]*dis[dst]  (one wave / edge) ----
__global__ void gcn_edge_scatter_kernel(const int* __restrict__ src,
                                        const int* __restrict__ dst,
                                        const float* __restrict__ dis,
                                        const float* __restrict__ H,
                                        float* __restrict__ agg, int E) {
  const int wave = (int)((blockIdx.x * (size_t)blockDim.x + threadIdx.x) >> 5);
  const int lane = threadIdx.x & 31;
  if (wave >= E) return;
  const int s = src[wave];
  const int d = dst[wave];
  const float nrm = dis[s] * dis[d];          // same-address loads broadcast in HW
  const float v = H[(size_t)s * HDIM + lane] * nrm;
  atomicAdd(&agg[(size_t)d * HDIM + lane], v);
}

// ---------- out = relu(agg + h*dis^2 (self loop) + bias), one wave / node ----------
__global__ void gcn_finalize_kernel(const float* __restrict__ agg,
                                    const float* __restrict__ H,
                                    const float* __restrict__ dis,
                                    const float* __restrict__ bias,
                                    float* __restrict__ out, int N) {
  const int node = (int)((blockIdx.x * (size_t)blockDim.x + threadIdx.x) >> 5);
  const int lane = threadIdx.x & 31;
  if (node >= N) return;
  const float dd = dis[node];
  const float v = agg[(size_t)node * HDIM + lane]
                + H[(size_t)node * HDIM + lane] * dd * dd
                + bias[lane];
  out[(size_t)node * HDIM + lane] = fmaxf(v, 0.0f);
}

extern "C" void kernel_launch(void* const* d_in, const int* in_sizes, int n_in,
                              void* d_out, int out_size, void* d_ws, size_t ws_size,
                              hipStream_t stream) {
  const float* x  = (const float*)d_in[0];
  const int*   ei = (const int*)d_in[1];
  const float* W1 = (const float*)d_in[2];
  const float* b1 = (const float*)d_in[3];
  const float* W2 = (const float*)d_in[4];
  const float* b2 = (const float*)d_in[5];
  float* out = (float*)d_out;

  const int N = in_sizes[0] / HDIM;
  const int E = in_sizes[1] / 2;
  const int* src = ei;        // edge_index[0, :]
  const int* dst = ei + E;    // edge_index[1, :]

  float* ws  = (float*)d_ws;
  float* dis = ws;                          // N floats (deg -> dis in place)
  float* h   = dis + N;                     // N*32
  float* agg = h + (size_t)N * HDIM;        // N*32
  float* y1  = agg + (size_t)N * HDIM;      // N*32

  const int NT = 256;
  // ----- normalization -----
  gcn_fill_kernel<<<(N + NT - 1) / NT, NT, 0, stream>>>(dis, 1.0f, N);          // self loop
  gcn_degree_kernel<<<(E + NT - 1) / NT, NT, 0, stream>>>(dst, dis, E);
  gcn_rsqrt_kernel<<<(N + NT - 1) / NT, NT, 0, stream>>>(dis, N);

  const int nTiles = (N + 15) >> 4;
  const int gemmBlocks = (nTiles + 3) / 4;            // 4 waves / block
  const int edgeBlocks = (E + 7) / 8;                 // 8 edges / 256-thread block
  const int nodeBlocks = (N + 7) / 8;                 // 8 nodes / 256-thread block

  // ----- layer 1 -----
  gcn_gemm_wmma_kernel<<<gemmBlocks, 128, 0, stream>>>(x, W1, h, N);
  gcn_fill_kernel<<<((size_t)N * HDIM + NT - 1) / NT, NT, 0, stream>>>(agg, 0.0f, N * HDIM);
  gcn_edge_scatter_kernel<<<edgeBlocks, 256, 0, stream>>>(src, dst, dis, h, agg, E);
  gcn_finalize_kernel<<<nodeBlocks, 256, 0, stream>>>(agg, h, dis, b1, y1, N);

  // ----- layer 2 -----
  gcn_gemm_wmma_kernel<<<gemmBlocks, 128, 0, stream>>>(y1, W2, h, N);
  gcn_fill_kernel<<<((size_t)N * HDIM + NT - 1) / NT, NT, 0, stream>>>(agg, 0.0f, N * HDIM);
  gcn_edge_scatter_kernel<<<edgeBlocks, 256, 0, stream>>>(src, dst, dis, h, agg, E);
  gcn_finalize_kernel<<<nodeBlocks, 256, 0, stream>>>(agg, h, dis, b2, out, N);
}